// Bilinear_Interpolation_84413287235759
// MI455X (gfx1250) — compile-verified
//
#include <hip/hip_runtime.h>
#include <hip/hip_bf16.h>

// ---------------------------------------------------------------------------
// Bilinear gather, MI455X (gfx1250). Memory-bound: ~1 GB moved, ~0.2 GFLOP.
// Two-phase: (1) pad+transpose feature map to channel-last in workspace,
// (2) one wave32 per point, b128 coalesced gather + NT b128 stores.
// ---------------------------------------------------------------------------

#define A_N   4096
#define TD_N  60
#define B_N   64
#define CE_N  128
#define H_N   100
#define W_N   100
#define HP_N  102
#define WP_N  102
#define P_N   (A_N * TD_N)            // 245760 points
#define OUT1_OFF ((size_t)P_N * CE_N) // 31,457,280 floats

typedef float __attribute__((ext_vector_type(4))) v4f;
typedef float __attribute__((ext_vector_type(2))) v2f;

// Exact unfused IEEE sequence to match JAX's elementwise ops bit-for-bit
// (floor/ceil pixel selection must not shift by fp-contraction).
__device__ __forceinline__ float map_coord(float s) {
    float t = __fadd_rn(s, 56.0f);
    t = __fdiv_rn(t, 112.0f);
    t = __fmul_rn(t, 100.0f);
    return __fadd_rn(t, 1.0f);
}

// ---------------------------------------------------------------------------
// Phase 1: fm (B, CE, 100, 100) -> fmT (B, 102, 102, CE), border = oom_val.
// One block = (b, y, 4 x-positions); thread = channel. Reads stream along x
// (lines reused by 4 consecutive x-tiles via L2), writes 512B coalesced.
// ---------------------------------------------------------------------------
__global__ __launch_bounds__(128) void pad_transpose_kernel(
    const float* __restrict__ fm, float* __restrict__ fmT,
    const float* __restrict__ oom_p)
{
    const float oom = *oom_p;
    const int bid = blockIdx.x;
    const int xt  = bid % 26;                 // 26 tiles of 4 cover 102
    const int y   = (bid / 26) % HP_N;
    const int b   = bid / (26 * HP_N);
    const int c   = threadIdx.x;

    const float* __restrict__ src =
        fm + ((size_t)(b * CE_N + c)) * (H_N * W_N);
    float* __restrict__ dstBase =
        fmT + ((size_t)(b * HP_N + y)) * WP_N * CE_N + c;

    const bool yin = (y >= 1) && (y <= H_N);
#pragma unroll
    for (int i = 0; i < 4; ++i) {
        const int x = xt * 4 + i;
        if (x < WP_N) {
            float v = oom;
            if (yin && (x >= 1) && (x <= W_N))
                v = src[(size_t)(y - 1) * W_N + (x - 1)];
            dstBase[(size_t)x * CE_N] = v;
        }
    }
}

// ---------------------------------------------------------------------------
// Phase 2: one wave32 per point; lane l owns channels 4l..4l+3 (v4f).
// 4 coalesced 512B row loads per point, NT stores (output is write-once,
// keep it out of L2 so the transposed map stays resident).
// ---------------------------------------------------------------------------
__global__ __launch_bounds__(256) void gather_kernel(
    const int* __restrict__ episode_idx, const float* __restrict__ sequence,
    const float* __restrict__ fmT, float* __restrict__ out)
{
    const int wid  = threadIdx.x >> 5;
    const int lane = threadIdx.x & 31;
    const int p    = blockIdx.x * 8 + wid;     // P_N % 8 == 0

    const int agent = p / TD_N;
    const int b     = episode_idx[agent];

    const float sx = sequence[2 * (size_t)p];
    const float sy = sequence[2 * (size_t)p + 1];
    const float x  = map_coord(sx);
    const float y  = map_coord(sy);

    const float x1 = fminf(fmaxf(floorf(x), 0.0f), 101.0f);
    const float y1 = fminf(fmaxf(floorf(y), 0.0f), 101.0f);
    const float x2 = fminf(fmaxf(ceilf(x),  0.0f), 101.0f);
    const float y2 = fminf(fmaxf(ceilf(y),  0.0f), 101.0f);
    const int xi1 = (int)x1, yi1 = (int)y1, xi2 = (int)x2, yi2 = (int)y2;

    // Reference's exact pairing: q11=(y1,x1), q21=(y2,x1), q12=(y1,x2), q22=(y2,x2)
    const float w11 = (x2 - x) * (y2 - y);
    const float w21 = (x  - x1) * (y2 - y);
    const float w12 = (x2 - x) * (y  - y1);
    const float w22 = (x  - x1) * (y  - y1);

    const size_t base = (size_t)b * (HP_N * WP_N * CE_N);
    const v4f* __restrict__ q11p =
        (const v4f*)(fmT + base + ((size_t)yi1 * WP_N + xi1) * CE_N) + lane;
    const v4f* __restrict__ q12p =
        (const v4f*)(fmT + base + ((size_t)yi1 * WP_N + xi2) * CE_N) + lane;
    const v4f* __restrict__ q21p =
        (const v4f*)(fmT + base + ((size_t)yi2 * WP_N + xi1) * CE_N) + lane;
    const v4f* __restrict__ q22p =
        (const v4f*)(fmT + base + ((size_t)yi2 * WP_N + xi2) * CE_N) + lane;

    const v4f q11 = *q11p;
    const v4f q21 = *q21p;
    const v4f q12 = *q12p;
    const v4f q22 = *q22p;

    const v4f r = q11 * w11 + q21 * w21 + q12 * w12 + q22 * w22;

    v4f* dst = (v4f*)(out + (size_t)p * CE_N) + lane;
    __builtin_nontemporal_store(r, dst);

    if (lane == 0) {
        v2f m; m.x = x; m.y = y;
        __builtin_nontemporal_store(m, (v2f*)(out + OUT1_OFF + 2 * (size_t)p));
    }
}

// ---------------------------------------------------------------------------
// Fallback (workspace too small for the 325 MiB transposed map): direct
// gather from (B, CE, H, W). One block per point, thread = channel.
// ---------------------------------------------------------------------------
__global__ __launch_bounds__(128) void gather_direct_kernel(
    const int* __restrict__ episode_idx, const float* __restrict__ sequence,
    const float* __restrict__ fm, const float* __restrict__ oom_p,
    float* __restrict__ out)
{
    const float oom = *oom_p;
    const int p = blockIdx.x;
    const int c = threadIdx.x;

    const int agent = p / TD_N;
    const int b     = episode_idx[agent];

    const float sx = sequence[2 * (size_t)p];
    const float sy = sequence[2 * (size_t)p + 1];
    const float x  = map_coord(sx);
    const float y  = map_coord(sy);

    const float x1 = fminf(fmaxf(floorf(x), 0.0f), 101.0f);
    const float y1 = fminf(fmaxf(floorf(y), 0.0f), 101.0f);
    const float x2 = fminf(fmaxf(ceilf(x),  0.0f), 101.0f);
    const float y2 = fminf(fmaxf(ceilf(y),  0.0f), 101.0f);
    const int xi1 = (int)x1, yi1 = (int)y1, xi2 = (int)x2, yi2 = (int)y2;

    const float w11 = (x2 - x) * (y2 - y);
    const float w21 = (x  - x1) * (y2 - y);
    const float w12 = (x2 - x) * (y  - y1);
    const float w22 = (x  - x1) * (y  - y1);

    const float* __restrict__ ch = fm + ((size_t)(b * CE_N + c)) * (H_N * W_N);
    auto fetch = [&](int yi, int xi) -> float {
        if (yi >= 1 && yi <= H_N && xi >= 1 && xi <= W_N)
            return ch[(size_t)(yi - 1) * W_N + (xi - 1)];
        return oom;
    };
    const float q11 = fetch(yi1, xi1);
    const float q12 = fetch(yi1, xi2);
    const float q21 = fetch(yi2, xi1);
    const float q22 = fetch(yi2, xi2);

    const float r = q11 * w11 + q21 * w21 + q12 * w12 + q22 * w22;
    __builtin_nontemporal_store(r, out + (size_t)p * CE_N + c);

    if (c == 0) {
        v2f m; m.x = x; m.y = y;
        __builtin_nontemporal_store(m, (v2f*)(out + OUT1_OFF + 2 * (size_t)p));
    }
}

// ---------------------------------------------------------------------------
extern "C" void kernel_launch(void* const* d_in, const int* in_sizes, int n_in,
                              void* d_out, int out_size, void* d_ws, size_t ws_size,
                              hipStream_t stream) {
    const int*   episode_idx = (const int*)  d_in[0];
    const float* sequence    = (const float*)d_in[1];
    const float* fm          = (const float*)d_in[2];
    const float* oom_val     = (const float*)d_in[3];
    float*       out         = (float*)d_out;

    const size_t need = (size_t)B_N * HP_N * WP_N * CE_N * sizeof(float); // ~325 MiB

    if (ws_size >= need) {
        float* fmT = (float*)d_ws;
        pad_transpose_kernel<<<B_N * HP_N * 26, 128, 0, stream>>>(fm, fmT, oom_val);
        gather_kernel<<<P_N / 8, 256, 0, stream>>>(episode_idx, sequence, fmT, out);
    } else {
        gather_direct_kernel<<<P_N, 128, 0, stream>>>(episode_idx, sequence, fm,
                                                      oom_val, out);
    }
}